// Model_21809843929577
// MI455X (gfx1250) — compile-verified
//
#include <hip/hip_runtime.h>

#define H1_   36
#define NG    144      // 4*H1
#define TLEN  2048
#define BATCH 2048
#define HSTR  40       // padded h row stride (floats)

typedef __attribute__((ext_vector_type(2))) float v2f;
typedef __attribute__((ext_vector_type(4))) float v4f;
typedef __attribute__((ext_vector_type(8))) float v8f;

__device__ __forceinline__ float tanh_f(float x) {
#if __has_builtin(__builtin_amdgcn_tanhf)
  return __builtin_amdgcn_tanhf(x);
#else
  float r;
  asm("v_tanh_f32 %0, %1" : "=v"(r) : "v"(x));
  return r;
#endif
}
__device__ __forceinline__ float sigm(float x) {
  // sigmoid(x) = 0.5*tanh(x/2) + 0.5   -> 1 v_tanh_f32 + fma
  return fmaf(0.5f, tanh_f(0.5f * x), 0.5f);
}
__device__ __forceinline__ v8f wmma4(v2f a, v2f b, v8f c) {
  // D = A(16x4 f32) * B(4x16 f32) + C(16x16 f32)
  return __builtin_amdgcn_wmma_f32_16x16x4_f32(false, a, false, b, (short)0, c,
                                               false, false);
}

__global__ __launch_bounds__(32) void lstm2_wmma_kernel(
    const float* __restrict__ x,
    const float* __restrict__ W_ih1, const float* __restrict__ W_hh1,
    const float* __restrict__ b_ih1, const float* __restrict__ b_hh1,
    const float* __restrict__ W_ih2, const float* __restrict__ W_hh2,
    const float* __restrict__ b_ih2, const float* __restrict__ b_hh2,
    float* __restrict__ out) {
  // Per-lane pre-swizzled WMMA B fragments for W_hh1^T (9 N-tiles x 9 K-chunks)
  __shared__ v2f  Bls[9][9][32];
  // W_ih2^T fragments (N padded 4 -> 16)
  __shared__ v2f  B2ls[9][32];
  __shared__ float gates[16 * NG];    // layer-1 pre-activation gates [seq][144]
  __shared__ float hbuf[16 * HSTR];   // layer-1 hidden state (pre-ReLU) [seq][36 pad 40]
  __shared__ float g2buf[16 * 16];    // layer-2 gate tile dump [seq][16]
  __shared__ float xbuf[16];          // x_t per sequence

  const int lane = threadIdx.x;
  const int lo = lane & 15;   // column / sequence sub-index
  const int hi = lane >> 4;   // K-halves selector in A/B layouts
  const int b0 = blockIdx.x * 16;

  // ---- Stage recurrent weights as WMMA B fragments ----
  // B[k][n] = W_hh1[n][k]; lane holds {B[k0][n], B[k0+1][n]}, n=16*ntile+lo,
  // k0 = 4*kchunk + 2*hi  (32-bit B-matrix 4x16 layout)
  for (int n = 0; n < 9; ++n)
    for (int k = 0; k < 9; ++k) {
      const int wrow = 16 * n + lo;
      const int kk = 4 * k + 2 * hi;
      Bls[n][k][lane] = *(const v2f*)(W_hh1 + wrow * H1_ + kk);
    }
  for (int k = 0; k < 9; ++k) {
    const int kk = 4 * k + 2 * hi;
    v2f v; v.x = 0.f; v.y = 0.f;
    if (lo < 4) v = *(const v2f*)(W_ih2 + lo * H1_ + kk);
    B2ls[k][lane] = v;
  }

  // Per-lane column constants: fold x*W_ih1 + (b_ih1+b_hh1) into WMMA C-init.
  float w1[9], bb1[9];
  for (int n = 0; n < 9; ++n) {
    const int g = 16 * n + lo;
    w1[n] = W_ih1[g];
    bb1[n] = b_ih1[g] + b_hh1[g];
  }
  float b2s[4], whh2[4];
  for (int g = 0; g < 4; ++g) {
    b2s[g] = b_ih2[g] + b_hh2[g];
    whh2[g] = W_hh2[g];
  }

  // h0 = 0
  for (int i = lane; i < 16 * HSTR; i += 32) hbuf[i] = 0.f;

  // c-state ownership: lane owns (seq = lo, j = 2u + hi), u = 0..17
  float c1[18];
#pragma unroll
  for (int u = 0; u < 18; ++u) c1[u] = 0.f;
  float h2 = 0.f, c2 = 0.f;

  float xnext = (hi == 0) ? x[(b0 + lo) * TLEN] : 0.f;

  v2f A1[9];  // A fragments of h1[t-1] (16x4 f32 layout, 9 K-chunks)

  auto build_A = [&]() {
#pragma unroll
    for (int k = 0; k < 9; ++k)
      A1[k] = *(const v2f*)(hbuf + lo * HSTR + 4 * k + 2 * hi);
  };

  // Layer-2 LSTM step for timestep tprev, consuming relu(h1[tprev]) = max(A1,0)
  auto layer2 = [&](int tprev) {
    v8f d2 = {};
#pragma unroll
    for (int k = 0; k < 9; ++k) {
      v2f a2;
      a2.x = fmaxf(A1[k].x, 0.f);
      a2.y = fmaxf(A1[k].y, 0.f);
      d2 = wmma4(a2, B2ls[k][lane], d2);
    }
#pragma unroll
    for (int v = 0; v < 8; ++v) g2buf[(v + 8 * hi) * 16 + lo] = d2[v];
    if (hi == 0) {
      const v4f g2 = *(const v4f*)(g2buf + lo * 16);
      const float pi = g2.x + b2s[0] + whh2[0] * h2;
      const float pf = g2.y + b2s[1] + whh2[1] * h2;
      const float pg = g2.z + b2s[2] + whh2[2] * h2;
      const float po = g2.w + b2s[3] + whh2[3] * h2;
      c2 = sigm(pf) * c2 + sigm(pi) * tanh_f(pg);
      h2 = sigm(po) * tanh_f(c2);
      out[(b0 + lo) * TLEN + tprev] = h2;
    }
  };

  for (int t = 0; t < TLEN; ++t) {
    if (hi == 0) xbuf[lo] = xnext;
    float xn2 = 0.f;
    if (hi == 0 && t + 1 < TLEN) xn2 = x[(b0 + lo) * TLEN + t + 1];  // prefetch

    build_A();               // A = h1[t-1]
    if (t > 0) layer2(t - 1);  // pipelined layer-2 for previous step

    // x_t per C-row: D-layout row of VGPR v is seq = v + 8*hi
    const v4f xva = *(const v4f*)(xbuf + 8 * hi);
    const v4f xvb = *(const v4f*)(xbuf + 8 * hi + 4);
    const float xv[8] = {xva.x, xva.y, xva.z, xva.w, xvb.x, xvb.y, xvb.z, xvb.w};

    // Layer-1 gates: 9 N-tiles x 9 K-chunks of V_WMMA_F32_16X16X4_F32
#pragma unroll
    for (int n = 0; n < 9; ++n) {
      v8f d;
#pragma unroll
      for (int v = 0; v < 8; ++v) d[v] = fmaf(xv[v], w1[n], bb1[n]);
#pragma unroll
      for (int k = 0; k < 9; ++k) d = wmma4(A1[k], Bls[n][k][lane], d);
#pragma unroll
      for (int v = 0; v < 8; ++v)
        gates[(v + 8 * hi) * NG + 16 * n + lo] = d[v];
    }

    // Layer-1 elementwise: PyTorch gate order i,f,g,o at j, j+36, j+72, j+108
#pragma unroll
    for (int u = 0; u < 18; ++u) {
      const int j = 2 * u + hi;
      const float* gp = gates + lo * NG + j;
      const float gi = gp[0], gf = gp[36], gg = gp[72], go = gp[108];
      const float cc = sigm(gf) * c1[u] + sigm(gi) * tanh_f(gg);
      c1[u] = cc;
      hbuf[lo * HSTR + j] = sigm(go) * tanh_f(cc);
    }
    xnext = xn2;
  }

  // Epilogue: layer-2 for the final timestep
  build_A();
  layer2(TLEN - 1);
}

extern "C" void kernel_launch(void* const* d_in, const int* in_sizes, int n_in,
                              void* d_out, int out_size, void* d_ws,
                              size_t ws_size, hipStream_t stream) {
  const float* x     = (const float*)d_in[0];
  const float* W_ih1 = (const float*)d_in[1];
  const float* W_hh1 = (const float*)d_in[2];
  const float* b_ih1 = (const float*)d_in[3];
  const float* b_hh1 = (const float*)d_in[4];
  const float* W_ih2 = (const float*)d_in[5];
  const float* W_hh2 = (const float*)d_in[6];
  const float* b_ih2 = (const float*)d_in[7];
  const float* b_hh2 = (const float*)d_in[8];
  float* out = (float*)d_out;

  dim3 grid(BATCH / 16);
  dim3 block(32);
  hipLaunchKernelGGL(lstm2_wmma_kernel, grid, block, 0, stream, x, W_ih1, W_hh1,
                     b_ih1, b_hh1, W_ih2, W_hh2, b_ih2, b_hh2, out);
}